// MNISTSPNet_10007273799860
// MI455X (gfx1250) — compile-verified
//
#include <hip/hip_runtime.h>
#include <hip/hip_bf16.h>

typedef float v2f __attribute__((ext_vector_type(2)));
typedef float v8f __attribute__((ext_vector_type(8)));

#define Nn 1048576
#define Ee 8388608
#define Hh 32
#define Gg 16384
#define Cc 10

__device__ __forceinline__ v8f wmma4(v2f a, v2f b, v8f c) {
    // D = A(16x4 f32) * B(4x16 f32) + C(16x16 f32)
    return __builtin_amdgcn_wmma_f32_16x16x4_f32(false, a, false, b, (short)0, c, false, false);
}

// ---------------- generic zero ----------------
__global__ void zero_kernel(float* __restrict__ p, long long n) {
    long long i = (long long)blockIdx.x * blockDim.x + threadIdx.x;
    if (i < n) p[i] = 0.0f;
}

// ---------------- node embedding: h = x @ W_emb.T + b_emb ----------------
__global__ void embed_kernel(const float* __restrict__ x, const float* __restrict__ W,
                             const float* __restrict__ b, float* __restrict__ h) {
    int node = blockIdx.x * 8 + (threadIdx.x >> 5);   // 8 nodes per 256-thread block
    int ch   = threadIdx.x & 31;
    const float* xr = x + (size_t)node * 5;
    const float* wr = W + ch * 5;
    float s = b[ch];
#pragma unroll
    for (int k = 0; k < 5; ++k) s += xr[k] * wr[k];
    h[(size_t)node * Hh + ch] = s;
}

// ---------------- edge scatter: agg[dst] += ew * h[src] ----------------
__global__ void scatter_kernel(const float* __restrict__ h, const int* __restrict__ src,
                               const int* __restrict__ dst, const float* __restrict__ ew,
                               float* __restrict__ agg) {
    long long t = (long long)blockIdx.x * blockDim.x + threadIdx.x;
    long long e = t >> 5;            // wave-uniform edge id
    int c = (int)(t & 31);
    int s = src[e];
    int d = dst[e];
    float v = h[(size_t)s * Hh + c] * ew[e];
    atomicAdd(&agg[(size_t)d * Hh + c], v);
}

// ---------------- conv GEMM: agg = agg@Wrel.T + h@Wroot.T + brel (in place) ----------------
// block = 256 threads = 8 waves; block covers 64 nodes x 32 channels
__global__ void conv_gemm_kernel(float* __restrict__ agg, const float* __restrict__ h,
                                 const float* __restrict__ Wrel, const float* __restrict__ Wroot,
                                 const float* __restrict__ brel) {
    int wave = threadIdx.x >> 5;
    int lane = threadIdx.x & 31;
    int node_base = blockIdx.x * 64 + (wave >> 1) * 16;
    int chan_base = (wave & 1) * 16;
    int M    = lane & 15;        // row index for A, col index for B/D
    int half = lane >> 4;        // K-pair select (A/B), row-halve select (D)
    int Ng   = chan_base + M;    // global output channel for this lane (B/D)

    const float* aggRow = agg + (size_t)(node_base + M) * Hh;
    const float* hRow   = h   + (size_t)(node_base + M) * Hh;
    const float* wrel   = Wrel  + Ng * Hh;   // B[k][j] = W[j][k]
    const float* wroot  = Wroot + Ng * Hh;

    v8f c = {};
#pragma unroll
    for (int kk = 0; kk < 8; ++kk) {
        int kb = kk * 4 + half * 2;
        v2f a1 = *(const v2f*)(aggRow + kb);
        v2f b1 = *(const v2f*)(wrel + kb);
        c = wmma4(a1, b1, c);
        v2f a2 = *(const v2f*)(hRow + kb);
        v2f b2 = *(const v2f*)(wroot + kb);
        c = wmma4(a2, b2, c);
    }
    float bias = brel[Ng];
#pragma unroll
    for (int r = 0; r < 8; ++r) {
        int row = node_base + half * 8 + r;
        agg[(size_t)row * Hh + Ng] = c[r] + bias;
    }
}

// ---------------- BN stats: per-channel sum / sumsq ----------------
__global__ void bn_stats_kernel(const float* __restrict__ v, float* __restrict__ stats) {
    __shared__ float s1[8][32];
    __shared__ float s2[8][32];
    int ch  = threadIdx.x & 31;
    int grp = threadIdx.x >> 5;
    float a = 0.0f, b = 0.0f;
    for (int row = blockIdx.x * 8 + grp; row < Nn; row += gridDim.x * 8) {
        float x = v[(size_t)row * Hh + ch];
        a += x; b += x * x;
    }
    s1[grp][ch] = a; s2[grp][ch] = b;
    __syncthreads();
    if (grp == 0) {
        float ta = 0.0f, tb = 0.0f;
#pragma unroll
        for (int g = 0; g < 8; ++g) { ta += s1[g][ch]; tb += s2[g][ch]; }
        atomicAdd(&stats[ch], ta);
        atomicAdd(&stats[32 + ch], tb);
    }
}

// ---------------- BN apply + ReLU (in place) ----------------
__global__ void bn_apply_kernel(float* __restrict__ v, const float* __restrict__ stats,
                                const float* __restrict__ gamma, const float* __restrict__ beta) {
    size_t idx = (size_t)blockIdx.x * blockDim.x + threadIdx.x;
    int ch = (int)(idx & 31);
    const float invN = 1.0f / (float)Nn;
    float mean = stats[ch] * invN;
    float var  = stats[32 + ch] * invN - mean * mean;
    float inv  = rsqrtf(var + 1e-5f);
    float y = (v[idx] - mean) * inv * gamma[ch] + beta[ch];
    v[idx] = y > 0.0f ? y : 0.0f;
}

// ---------------- segment max pool (values >= 0 post-ReLU -> uint atomicMax valid) ----------------
__global__ void pool_max_kernel(const float* __restrict__ h, const int* __restrict__ batch,
                                unsigned int* __restrict__ pooled) {
    size_t idx = (size_t)blockIdx.x * blockDim.x + threadIdx.x;
    int node = (int)(idx >> 5);   // wave-uniform
    int ch   = (int)(idx & 31);
    int g = batch[node];
    atomicMax(&pooled[(size_t)g * Hh + ch], __float_as_uint(h[idx]));
}

// ---------------- MLP1: hidden = relu(pooled @ W1.T + b1)  [G x 64], WMMA ----------------
// block = 256 threads = 8 waves; block covers 32 graphs x 64 channels
__global__ void mlp1_kernel(const float* __restrict__ pooled, const float* __restrict__ W1,
                            const float* __restrict__ b1, float* __restrict__ hidden) {
    int wave = threadIdx.x >> 5;
    int lane = threadIdx.x & 31;
    int g_base    = blockIdx.x * 32 + (wave >> 2) * 16;
    int chan_base = (wave & 3) * 16;
    int M    = lane & 15;
    int half = lane >> 4;
    int Ng   = chan_base + M;

    const float* pRow = pooled + (size_t)(g_base + M) * Hh;
    const float* wRow = W1 + Ng * Hh;   // W1 is [64,32] row-major; B[k][j] = W1[j][k]

    v8f c = {};
#pragma unroll
    for (int kk = 0; kk < 8; ++kk) {
        int kb = kk * 4 + half * 2;
        v2f a = *(const v2f*)(pRow + kb);
        v2f b = *(const v2f*)(wRow + kb);
        c = wmma4(a, b, c);
    }
    float bias = b1[Ng];
#pragma unroll
    for (int r = 0; r < 8; ++r) {
        int row = g_base + half * 8 + r;
        float y = c[r] + bias;
        hidden[(size_t)row * 64 + Ng] = y > 0.0f ? y : 0.0f;
    }
}

// ---------------- MLP2: out = hidden @ W2.T + b2  [G x 10], tiny, scalar ----------------
__global__ void mlp2_kernel(const float* __restrict__ hidden, const float* __restrict__ W2,
                            const float* __restrict__ b2, float* __restrict__ out) {
    int idx = blockIdx.x * blockDim.x + threadIdx.x;
    if (idx >= Gg * Cc) return;
    int g = idx / Cc;
    int c = idx % Cc;
    const float* hr = hidden + (size_t)g * 64;
    const float* wr = W2 + c * 64;
    float s = b2[c];
#pragma unroll
    for (int k = 0; k < 64; ++k) s += hr[k] * wr[k];
    out[idx] = s;
}

extern "C" void kernel_launch(void* const* d_in, const int* in_sizes, int n_in,
                              void* d_out, int out_size, void* d_ws, size_t ws_size,
                              hipStream_t stream) {
    const float* x      = (const float*)d_in[0];
    const int*   ei     = (const int*)d_in[1];
    const float* ea     = (const float*)d_in[2];
    const int*   batch  = (const int*)d_in[3];
    const float* W_emb  = (const float*)d_in[4];
    const float* b_emb  = (const float*)d_in[5];
    const float* W_rel  = (const float*)d_in[6];   // [2][32][32]
    const float* b_rel  = (const float*)d_in[7];   // [2][32]
    const float* W_root = (const float*)d_in[8];   // [2][32][32]
    const float* gamma  = (const float*)d_in[9];   // [2][32]
    const float* beta   = (const float*)d_in[10];  // [2][32]
    const float* W1     = (const float*)d_in[11];  // [64,32]
    const float* b1     = (const float*)d_in[12];  // [64]
    const float* W2     = (const float*)d_in[13];  // [10,64]
    const float* b2     = (const float*)d_in[14];  // [10]
    float* out = (float*)d_out;

    const int* src = ei;
    const int* dst = ei + Ee;

    float* ws     = (float*)d_ws;
    float* hA     = ws;                       // N*32
    float* hB     = hA + (size_t)Nn * Hh;     // N*32
    float* stats  = hB + (size_t)Nn * Hh;     // 64
    float* pooled = stats + 64;               // G*32
    float* hidden = pooled + (size_t)Gg * Hh; // G*64

    const long long NH = (long long)Nn * Hh;

    // node embedding -> hA
    embed_kernel<<<Nn / 8, 256, 0, stream>>>(x, W_emb, b_emb, hA);

    float* hcur = hA;
    float* hagg = hB;
    for (int l = 0; l < 2; ++l) {
        zero_kernel<<<(unsigned)(NH / 256), 256, 0, stream>>>(hagg, NH);
        zero_kernel<<<1, 64, 0, stream>>>(stats, 64);
        scatter_kernel<<<(unsigned)((long long)Ee * 32 / 256), 256, 0, stream>>>(hcur, src, dst, ea, hagg);
        conv_gemm_kernel<<<Nn / 64, 256, 0, stream>>>(hagg, hcur,
                                                      W_rel + l * Hh * Hh, W_root + l * Hh * Hh,
                                                      b_rel + l * Hh);
        bn_stats_kernel<<<1024, 256, 0, stream>>>(hagg, stats);
        bn_apply_kernel<<<(unsigned)(NH / 256), 256, 0, stream>>>(hagg, stats,
                                                                  gamma + l * Hh, beta + l * Hh);
        float* t = hcur; hcur = hagg; hagg = t;   // output of layer becomes input
    }

    // global max pool
    zero_kernel<<<(Gg * Hh) / 256, 256, 0, stream>>>(pooled, (long long)Gg * Hh);
    pool_max_kernel<<<(unsigned)(NH / 256), 256, 0, stream>>>(hcur, batch, (unsigned int*)pooled);

    // MLP head
    mlp1_kernel<<<Gg / 32, 256, 0, stream>>>(pooled, W1, b1, hidden);
    mlp2_kernel<<<(Gg * Cc + 255) / 256, 256, 0, stream>>>(hidden, W2, b2, out);
}